// LocalFeatureExtractor_70007966925259
// MI455X (gfx1250) — compile-verified
//
#include <hip/hip_runtime.h>
#include <math.h>

// ---------------------------------------------------------------------------
// LocalFeatureExtractor for MI455X (gfx1250, wave32, WMMA).
// All GEMMs use V_WMMA_F32_16X16X4_F32 (fp32 A/B/C, 16x16 tiles, K=4) to keep
// the reference's fp32 numerics. Workload is bandwidth/selection bound
// (~300MB traffic @ 23.3TB/s), so fp32 WMMA is the right precision choice.
// ---------------------------------------------------------------------------

typedef __attribute__((ext_vector_type(2))) float v2f;
typedef __attribute__((ext_vector_type(8))) float v8f;

__device__ __forceinline__ v8f wmma4(v2f a, v2f b, v8f c) {
  // D = A(16x4) * B(4x16) + C(16x16), fp32
  return __builtin_amdgcn_wmma_f32_16x16x4_f32(false, a, false, b, (short)0, c,
                                               false, false);
}
__device__ __forceinline__ v8f z8() {
  v8f z = {0.f, 0.f, 0.f, 0.f, 0.f, 0.f, 0.f, 0.f};
  return z;
}

constexpr int B_ = 2, C_ = 128, N_ = 8192, P_ = 1024, NS_ = 32;
constexpr int CO = 256, CIN = 131, CINP = 132;   // padded K for WMMA
constexpr float RADIUS2 = 0.3f * 0.3f, HMIN = -0.2f, HMAX = 0.2f;
constexpr float EPS = 1e-5f;

// workspace layout (bytes)
constexpr size_t OFF_IDX  = 0;                                   // 65536 int
constexpr size_t OFF_WP   = 262144;                              // 256*132 f32
constexpr size_t OFF_PAR  = OFF_WP + (size_t)CO * CINP * 4;      // 6*256 f32
constexpr size_t OFF_H    = 1u << 20;                            // 64 MB
constexpr size_t OFF_Y    = OFF_H + (size_t)2048 * CO * NS_ * 4; // 64 MB
constexpr size_t OFF_POOL = OFF_Y + (size_t)2048 * CO * NS_ * 4; // 2 MB
constexpr size_t OFF_ZR   = OFF_POOL + (size_t)B_ * CO * P_ * 4; // 2 MB

// ---------------------------------------------------------------------------
// K0: pad mlp_w (256x131) -> (256x132) so B-fragments are aligned float2
// ---------------------------------------------------------------------------
__global__ void k_pack(const float* __restrict__ w, float* __restrict__ wp) {
  int i = blockIdx.x * 256 + threadIdx.x;
  if (i >= CO * CINP) return;
  int o = i / CINP, c = i % CINP;
  wp[i] = (c < CIN) ? w[o * CIN + c] : 0.0f;
}

// ---------------------------------------------------------------------------
// K1: cylinder query. One wave32 per seed; stable first-32-valid selection
// using ballot/popc; pad with first valid index (or 0 if none) - matches the
// reference's stable-argsort + count-fill semantics.
// ---------------------------------------------------------------------------
__global__ void __launch_bounds__(256)
k_select(const float* __restrict__ xyz, const float* __restrict__ seed,
         const float* __restrict__ rot, int* __restrict__ idx) {
  int wave = threadIdx.x >> 5, lane = threadIdx.x & 31;
  int g = blockIdx.x * 8 + wave;
  int b = g >> 10, p = g & 1023;
  const float* sd = seed + (size_t)(b * P_ + p) * 3;
  const float* R = rot + (size_t)(b * P_ + p) * 9;
  float sx = sd[0], sy = sd[1], sz = sd[2];
  float r00 = R[0], r01 = R[1], r02 = R[2];
  float r10 = R[3], r11 = R[4], r12 = R[5];
  float r20 = R[6], r21 = R[7], r22 = R[8];
  const float* X = xyz + (size_t)b * N_ * 3;
  int count = 0, first = 0;
  for (int n0 = 0; n0 < N_; n0 += 32) {
    int n = n0 + lane;
    float d0 = X[n * 3 + 0] - sx, d1 = X[n * 3 + 1] - sy, d2 = X[n * 3 + 2] - sz;
    float t0 = d0 * r00 + d1 * r10 + d2 * r20;
    float t1 = d0 * r01 + d1 * r11 + d2 * r21;
    float t2 = d0 * r02 + d1 * r12 + d2 * r22;
    bool valid = (t1 * t1 + t2 * t2 < RADIUS2) && (t0 > HMIN) && (t0 < HMAX);
    unsigned mask = (unsigned)__ballot(valid);
    if (count == 0 && mask) first = n0 + __ffs(mask) - 1;
    if (valid) {
      int pos = count + __popc(mask & ((1u << lane) - 1u));
      if (pos < NS_) idx[g * NS_ + pos] = n;
    }
    count += __popc(mask);
    if (count >= NS_) break;
  }
  if (count < NS_ && lane >= count) idx[g * NS_ + lane] = first;
}

// ---------------------------------------------------------------------------
// K2: gather (rotated coords + features) -> LDS, then SharedMLP GEMM
// h[s][o] = nf[s][0:131] * W[o][0:131]^T + b[o].  WMMA tiles: M=s, N=o, K=c.
// Output layout h_raw[(g*256+o)*32+s].
// ---------------------------------------------------------------------------
__global__ void __launch_bounds__(256)
k_mlp(const float* __restrict__ feat, const float* __restrict__ xyz,
      const float* __restrict__ seed, const float* __restrict__ rot,
      const int* __restrict__ idx, const float* __restrict__ wp,
      const float* __restrict__ bias, float* __restrict__ hraw) {
  __shared__ __align__(16) float nf[NS_ * CINP];  // [s][c]
  int g = blockIdx.x, b = g >> 10, p = g & 1023;
  const float* sd = seed + (size_t)(b * P_ + p) * 3;
  const float* R = rot + (size_t)(b * P_ + p) * 9;
  for (int i = threadIdx.x; i < NS_ * CINP; i += 256) {
    int s = i / CINP, c = i % CINP;
    int n = idx[g * NS_ + s];
    float v;
    if (c < 3) {
      const float* pt = xyz + ((size_t)b * N_ + n) * 3;
      float d0 = pt[0] - sd[0], d1 = pt[1] - sd[1], d2 = pt[2] - sd[2];
      v = d0 * R[0 * 3 + c] + d1 * R[1 * 3 + c] + d2 * R[2 * 3 + c];
    } else if (c < CIN) {
      v = feat[((size_t)b * C_ + (c - 3)) * N_ + n];
    } else {
      v = 0.0f;
    }
    nf[i] = v;
  }
  __syncthreads();
  int wave = threadIdx.x >> 5, lane = threadIdx.x & 31;
  int half = lane >> 4, l15 = lane & 15;
  for (int t = 0; t < 4; ++t) {
    int nt = wave * 2 + (t & 1);  // output-channel tile 0..15
    int mt = t >> 1;              // sample tile 0..1
    int ocol = nt * 16 + l15;
    v8f acc = z8();
    for (int k = 0; k < CINP; k += 4) {
      int kk = k + 2 * half;
      v2f a;
      a.x = nf[(mt * 16 + l15) * CINP + kk];
      a.y = nf[(mt * 16 + l15) * CINP + kk + 1];
      v2f bb = *reinterpret_cast<const v2f*>(wp + (size_t)ocol * CINP + kk);
      acc = wmma4(a, bb, acc);
    }
    float bo = bias[ocol];
    for (int r = 0; r < 8; ++r) {
      int s = mt * 16 + r + 8 * half;
      hraw[((size_t)g * CO + ocol) * NS_ + s] = acc[r] + bo;
    }
  }
}

// ---------------------------------------------------------------------------
// BN stats (deterministic, atomic-free): buf layout [m][c][s], m=2048, s=32.
// Produces per-channel scale/shift: y_bn = y*sc + sh.
// ---------------------------------------------------------------------------
__global__ void __launch_bounds__(256)
k_bnstats_mcs(const float* __restrict__ buf, const float* __restrict__ gam,
              const float* __restrict__ bet, float* __restrict__ sc,
              float* __restrict__ sh) {
  int c = blockIdx.x, t = threadIdx.x;
  float s = 0.f, q = 0.f;
  for (int m = t; m < 2048; m += 256) {
    const float* pb = buf + ((size_t)m * CO + c) * NS_;
    for (int i = 0; i < NS_; ++i) { float v = pb[i]; s += v; q += v * v; }
  }
  __shared__ float ss[256], qq[256];
  ss[t] = s; qq[t] = q;
  __syncthreads();
  for (int o = 128; o > 0; o >>= 1) {
    if (t < o) { ss[t] += ss[t + o]; qq[t] += qq[t + o]; }
    __syncthreads();
  }
  if (t == 0) {
    float mean = ss[0] * (1.0f / 65536.0f);
    float var = qq[0] * (1.0f / 65536.0f) - mean * mean;
    float k = gam[c] * rsqrtf(var + EPS);
    sc[c] = k;
    sh[c] = bet[c] - mean * k;
  }
}

// buf layout [b][c][p], count = B*P = 2048
__global__ void __launch_bounds__(256)
k_bnstats_bcp(const float* __restrict__ buf, const float* __restrict__ gam,
              const float* __restrict__ bet, float* __restrict__ sc,
              float* __restrict__ sh) {
  int c = blockIdx.x, t = threadIdx.x;
  float s = 0.f, q = 0.f;
  for (int i = t; i < B_ * P_; i += 256) {
    int b = i >> 10, p = i & 1023;
    float v = buf[((size_t)b * CO + c) * P_ + p];
    s += v; q += v * v;
  }
  __shared__ float ss[256], qq[256];
  ss[t] = s; qq[t] = q;
  __syncthreads();
  for (int o = 128; o > 0; o >>= 1) {
    if (t < o) { ss[t] += ss[t + o]; qq[t] += qq[t + o]; }
    __syncthreads();
  }
  if (t == 0) {
    float mean = ss[0] * (1.0f / 2048.0f);
    float var = qq[0] * (1.0f / 2048.0f) - mean * mean;
    float k = gam[c] * rsqrtf(var + EPS);
    sc[c] = k;
    sh[c] = bet[c] - mean * k;
  }
}

// ---------------------------------------------------------------------------
// K4: local attention per group (block = one group, 8 waves).
// x = relu(bn1(h));  G = q_w@x (64x32);  S = G^T G (32x32);  att = softmax;
// xv = v_w@x + vb;  u = x - xv@att;  y_raw = t_w@u + tb.
// ---------------------------------------------------------------------------
__global__ void __launch_bounds__(256)
k_attn(const float* __restrict__ hraw, const float* __restrict__ sc1,
       const float* __restrict__ sh1, const float* __restrict__ qw,
       const float* __restrict__ vw, const float* __restrict__ vb,
       const float* __restrict__ tw, const float* __restrict__ tb,
       float* __restrict__ yraw) {
  __shared__ __align__(16) float xs[CO * NS_];  // x, later u = x - xr (32KB)
  __shared__ __align__(16) float vs[CO * NS_];  // xv                 (32KB)
  __shared__ __align__(16) float Gs[64 * NS_];  // G = q_w @ x        (8KB)
  __shared__ __align__(16) float Ss[32 * 33];   // scores / att       (4.2KB)
  int g = blockIdx.x;
  int wave = threadIdx.x >> 5, lane = threadIdx.x & 31;
  int half = lane >> 4, l15 = lane & 15;

  // load + bn1 + relu -> x[c][s]
  for (int i = threadIdx.x; i < CO * NS_; i += 256) {
    int c = i >> 5;
    float v = hraw[(size_t)g * CO * NS_ + i] * sc1[c] + sh1[c];
    xs[i] = v > 0.f ? v : 0.f;
  }
  __syncthreads();

  // G = q_w(64x256) @ x(256x32): 8 tiles, one per wave
  {
    int mt = wave >> 1, nt = wave & 1;
    int mrow = mt * 16 + l15, ncol = nt * 16 + l15;
    v8f acc = z8();
    for (int k = 0; k < CO; k += 4) {
      int kk = k + 2 * half;
      v2f a = *reinterpret_cast<const v2f*>(qw + (size_t)mrow * CO + kk);
      v2f bb;
      bb.x = xs[kk * NS_ + ncol];
      bb.y = xs[(kk + 1) * NS_ + ncol];
      acc = wmma4(a, bb, acc);
    }
    for (int r = 0; r < 8; ++r) Gs[(mt * 16 + r + 8 * half) * NS_ + ncol] = acc[r];
  }
  __syncthreads();

  // S = G^T(32x64) @ G(64x32): 4 tiles on waves 0..3
  if (wave < 4) {
    int mt = wave >> 1, nt = wave & 1;
    v8f acc = z8();
    for (int k = 0; k < 64; k += 4) {
      int kk = k + 2 * half;
      v2f a, bb;
      a.x = Gs[kk * NS_ + mt * 16 + l15];
      a.y = Gs[(kk + 1) * NS_ + mt * 16 + l15];
      bb.x = Gs[kk * NS_ + nt * 16 + l15];
      bb.y = Gs[(kk + 1) * NS_ + nt * 16 + l15];
      acc = wmma4(a, bb, acc);
    }
    for (int r = 0; r < 8; ++r)
      Ss[(mt * 16 + r + 8 * half) * 33 + nt * 16 + l15] = acc[r];
  }
  __syncthreads();

  // row softmax; reference then divides by (1e-9 + rowsum) with rowsum==1
  if (threadIdx.x < 32) {
    int r = threadIdx.x;
    float mx = -1e30f;
    for (int j = 0; j < 32; ++j) mx = fmaxf(mx, Ss[r * 33 + j]);
    float sum = 0.f;
    for (int j = 0; j < 32; ++j) {
      float e = expf(Ss[r * 33 + j] - mx);
      Ss[r * 33 + j] = e;
      sum += e;
    }
    float inv = 1.0f / (sum * (1.0f + 1e-9f));
    for (int j = 0; j < 32; ++j) Ss[r * 33 + j] *= inv;
  }
  __syncthreads();

  // xv = v_w(256x256) @ x + vb: 32 tiles, 4 per wave
  for (int t = 0; t < 4; ++t) {
    int mt = wave * 2 + (t >> 1), nt = t & 1;
    int mrow = mt * 16 + l15, ncol = nt * 16 + l15;
    v8f acc = z8();
    for (int k = 0; k < CO; k += 4) {
      int kk = k + 2 * half;
      v2f a = *reinterpret_cast<const v2f*>(vw + (size_t)mrow * CO + kk);
      v2f bb;
      bb.x = xs[kk * NS_ + ncol];
      bb.y = xs[(kk + 1) * NS_ + ncol];
      acc = wmma4(a, bb, acc);
    }
    for (int r = 0; r < 8; ++r) {
      int o = mt * 16 + r + 8 * half;
      vs[o * NS_ + ncol] = acc[r] + vb[o];
    }
  }
  __syncthreads();

  // xr = xv(256x32) @ att(32x32); u = x - xr written in place into xs
  for (int t = 0; t < 4; ++t) {
    int mt = wave * 2 + (t >> 1), nt = t & 1;
    int mrow = mt * 16 + l15, ncol = nt * 16 + l15;
    v8f acc = z8();
    for (int k = 0; k < NS_; k += 4) {
      int kk = k + 2 * half;
      v2f a, bb;
      a.x = vs[mrow * NS_ + kk];
      a.y = vs[mrow * NS_ + kk + 1];
      bb.x = Ss[kk * 33 + ncol];
      bb.y = Ss[(kk + 1) * 33 + ncol];
      acc = wmma4(a, bb, acc);
    }
    for (int r = 0; r < 8; ++r) {
      int o = mt * 16 + r + 8 * half;
      xs[o * NS_ + ncol] = xs[o * NS_ + ncol] - acc[r];
    }
  }
  __syncthreads();

  // y = t_w(256x256) @ u + tb -> y_raw
  for (int t = 0; t < 4; ++t) {
    int mt = wave * 2 + (t >> 1), nt = t & 1;
    int mrow = mt * 16 + l15, ncol = nt * 16 + l15;
    v8f acc = z8();
    for (int k = 0; k < CO; k += 4) {
      int kk = k + 2 * half;
      v2f a = *reinterpret_cast<const v2f*>(tw + (size_t)mrow * CO + kk);
      v2f bb;
      bb.x = xs[kk * NS_ + ncol];
      bb.y = xs[(kk + 1) * NS_ + ncol];
      acc = wmma4(a, bb, acc);
    }
    for (int r = 0; r < 8; ++r) {
      int o = mt * 16 + r + 8 * half;
      yraw[((size_t)g * CO + o) * NS_ + ncol] = acc[r] + tb[o];
    }
  }
}

// ---------------------------------------------------------------------------
// K6: x_final = relu(bn1(h)) + relu(bn2(y)); max-pool over samples -> pooled
// ---------------------------------------------------------------------------
__global__ void __launch_bounds__(256)
k_combine(const float* __restrict__ hraw, const float* __restrict__ yraw,
          const float* __restrict__ sc1, const float* __restrict__ sh1,
          const float* __restrict__ sc2, const float* __restrict__ sh2,
          float* __restrict__ pooled) {
  int g = blockIdx.x, c = threadIdx.x;
  int b = g >> 10, p = g & 1023;
  size_t base = ((size_t)g * CO + c) * NS_;
  float a1 = sc1[c], b1 = sh1[c], a2 = sc2[c], b2 = sh2[c];
  float mx = -1e30f;
  for (int s = 0; s < NS_; ++s) {
    float x = hraw[base + s] * a1 + b1;
    x = x > 0.f ? x : 0.f;
    float y = yraw[base + s] * a2 + b2;
    y = y > 0.f ? y : 0.f;
    mx = fmaxf(mx, x + y);
  }
  pooled[((size_t)b * CO + c) * P_ + p] = mx;
}

// ---------------------------------------------------------------------------
// K7: z_relu = relu(adj_w(256x256) @ pooled(256x2048) + adj_b)
// One block per 16-column strip; 16 M-tiles, 2 per wave.
// ---------------------------------------------------------------------------
__global__ void __launch_bounds__(256)
k_adjust(const float* __restrict__ pooled, const float* __restrict__ aw,
         const float* __restrict__ ab, float* __restrict__ zr) {
  __shared__ __align__(16) float Bp[CO * 16];  // [c][p] strip (16KB)
  int bx = blockIdx.x;
  int b = bx >> 6;            // 64 strips of 16 per batch
  int p0 = (bx & 63) * 16;
  for (int i = threadIdx.x; i < CO * 16; i += 256) {
    int c = i >> 4, pc = i & 15;
    Bp[i] = pooled[((size_t)b * CO + c) * P_ + p0 + pc];
  }
  __syncthreads();
  int wave = threadIdx.x >> 5, lane = threadIdx.x & 31;
  int half = lane >> 4, l15 = lane & 15;
  for (int t = 0; t < 2; ++t) {
    int mt = wave * 2 + t;
    int mrow = mt * 16 + l15;
    v8f acc = z8();
    for (int k = 0; k < CO; k += 4) {
      int kk = k + 2 * half;
      v2f a = *reinterpret_cast<const v2f*>(aw + (size_t)mrow * CO + kk);
      v2f bb;
      bb.x = Bp[kk * 16 + l15];
      bb.y = Bp[(kk + 1) * 16 + l15];
      acc = wmma4(a, bb, acc);
    }
    for (int r = 0; r < 8; ++r) {
      int o = mt * 16 + r + 8 * half;
      float v = acc[r] + ab[o];
      zr[((size_t)b * CO + o) * P_ + p0 + l15] = v > 0.f ? v : 0.f;
    }
  }
}

// K9: final BN apply -> output (B,256,P)
__global__ void __launch_bounds__(256)
k_apply(const float* __restrict__ zr, const float* __restrict__ sc,
        const float* __restrict__ sh, float* __restrict__ out) {
  int i = blockIdx.x * 256 + threadIdx.x;  // 524288 total
  int c = (i >> 10) & 255;
  out[i] = zr[i] * sc[c] + sh[c];
}

// ---------------------------------------------------------------------------
extern "C" void kernel_launch(void* const* d_in, const int* in_sizes, int n_in,
                              void* d_out, int out_size, void* d_ws,
                              size_t ws_size, hipStream_t stream) {
  (void)in_sizes; (void)n_in; (void)out_size; (void)ws_size;
  const float* feat     = (const float*)d_in[0];
  const float* xyz      = (const float*)d_in[1];
  const float* seed     = (const float*)d_in[2];
  const float* rot      = (const float*)d_in[3];
  const float* mlp_w    = (const float*)d_in[4];
  const float* mlp_b    = (const float*)d_in[5];
  const float* mlp_g    = (const float*)d_in[6];
  const float* mlp_beta = (const float*)d_in[7];
  const float* q_w      = (const float*)d_in[8];
  const float* v_w      = (const float*)d_in[9];
  const float* v_b      = (const float*)d_in[10];
  const float* t_w      = (const float*)d_in[11];
  const float* t_b      = (const float*)d_in[12];
  const float* att_g    = (const float*)d_in[13];
  const float* att_beta = (const float*)d_in[14];
  const float* adj_w    = (const float*)d_in[15];
  const float* adj_b    = (const float*)d_in[16];
  const float* adj_g    = (const float*)d_in[17];
  const float* adj_beta = (const float*)d_in[18];

  char* ws = (char*)d_ws;
  int*   idx    = (int*)(ws + OFF_IDX);
  float* wp     = (float*)(ws + OFF_WP);
  float* par    = (float*)(ws + OFF_PAR);
  float* hraw   = (float*)(ws + OFF_H);
  float* yraw   = (float*)(ws + OFF_Y);
  float* pooled = (float*)(ws + OFF_POOL);
  float* zr     = (float*)(ws + OFF_ZR);
  float* sc1 = par, *sh1 = par + 256, *sc2 = par + 512;
  float* sh2 = par + 768, *sc3 = par + 1024, *sh3 = par + 1280;

  k_pack<<<(CO * CINP + 255) / 256, 256, 0, stream>>>(mlp_w, wp);
  k_select<<<256, 256, 0, stream>>>(xyz, seed, rot, idx);
  k_mlp<<<2048, 256, 0, stream>>>(feat, xyz, seed, rot, idx, wp, mlp_b, hraw);
  k_bnstats_mcs<<<256, 256, 0, stream>>>(hraw, mlp_g, mlp_beta, sc1, sh1);
  k_attn<<<2048, 256, 0, stream>>>(hraw, sc1, sh1, q_w, v_w, v_b, t_w, t_b, yraw);
  k_bnstats_mcs<<<256, 256, 0, stream>>>(yraw, att_g, att_beta, sc2, sh2);
  k_combine<<<2048, 256, 0, stream>>>(hraw, yraw, sc1, sh1, sc2, sh2, pooled);
  k_adjust<<<128, 256, 0, stream>>>(pooled, adj_w, adj_b, zr);
  k_bnstats_bcp<<<256, 256, 0, stream>>>(zr, adj_g, adj_beta, sc3, sh3);
  k_apply<<<2048, 256, 0, stream>>>(zr, sc3, sh3, (float*)d_out);
}